// HybridPerceptionCortex_68401649156463
// MI455X (gfx1250) — compile-verified
//
#include <hip/hip_runtime.h>
#include <hip/hip_bf16.h>

typedef float v2f __attribute__((ext_vector_type(2)));
typedef float v8f __attribute__((ext_vector_type(8)));

#define BATCH 4096
#define NN    4096
#define FEAT  512
#define MAPH  256
#define MAPW  256
#define MTOT  (MAPH*MAPW)     // 65536
#define LRATE 0.005f
#define THRESH 1.0f

// ws layout (floats)
#define WS_X       0            // 4096 : column sums of sensory
#define WS_CUR     4096         // 512  : current
#define WS_OFF     4608         // 512  : out_ff
#define WS_FEAT    5120         // 512  : feat
#define WS_D2      5632         // 65536: d2_0
#define WS_ZERON   5120         // floats to zero (x + current + out_ff)

// ---------------- K0: zero accumulators ----------------
__global__ void k_zero(float* ws, int n) {
    int i = blockIdx.x * blockDim.x + threadIdx.x;
    if (i < n) ws[i] = 0.0f;
}

// ---------------- K1: column sum of sensory [BATCH,NN] ----------------
__global__ void k_colsum(const float* __restrict__ S, float* __restrict__ ws) {
    int col = blockIdx.x * 256 + threadIdx.x;          // 16 blocks.x * 256
    int r0  = blockIdx.y * 128;                        // 32 blocks.y * 128 rows
    float acc = 0.0f;
    const float* p = S + (size_t)r0 * NN + col;
    #pragma unroll 4
    for (int r = 0; r < 128; ++r) {
        acc += p[(size_t)r * NN];
    }
    atomicAdd(&ws[WS_X + col], acc);
}

// ---------------- K2: current[f] = mean(x) @ W_in ----------------
__global__ void k_gemv_in(const float* __restrict__ Win, float* __restrict__ ws) {
    int f  = blockIdx.x * 256 + threadIdx.x;           // 2 blocks.x
    int n0 = blockIdx.y * 256;                         // 16 blocks.y
    const float scale = 1.0f / (float)BATCH;
    float acc = 0.0f;
    #pragma unroll 4
    for (int n = 0; n < 256; ++n) {
        float xv = ws[WS_X + n0 + n] * scale;
        acc += xv * Win[(size_t)(n0 + n) * FEAT + f];
    }
    atomicAdd(&ws[WS_CUR + f], acc);
}

// ---------------- K3: out_ff[j] = spikes @ W_out ----------------
__global__ void k_gemv_out(const float* __restrict__ Wout, float* __restrict__ ws) {
    int j  = blockIdx.x * 256 + threadIdx.x;           // 2 blocks.x
    int f0 = blockIdx.y * 128;                         // 4 blocks.y
    float acc = 0.0f;
    #pragma unroll 4
    for (int f = 0; f < 128; ++f) {
        float cur = ws[WS_CUR + f0 + f];
        if (cur >= THRESH) acc += Wout[(size_t)(f0 + f) * FEAT + j];
    }
    atomicAdd(&ws[WS_OFF + j], acc);
}

// ---------------- K4: feat[i] = relu(proj_W[i,:] . out_ff + b[i]) ----------------
__global__ void k_feat(const float* __restrict__ P, const float* __restrict__ b,
                       float* __restrict__ ws) {
    int wave = threadIdx.x >> 5;
    int lane = threadIdx.x & 31;
    int i = blockIdx.x * 8 + wave;                     // 64 blocks * 8 waves = 512 rows
    float acc = 0.0f;
    #pragma unroll
    for (int t = 0; t < 4; ++t) {
        int idx = t * 128 + lane * 4;
        float4 p = *(const float4*)&P[(size_t)i * FEAT + idx];
        float4 o = *(const float4*)&ws[WS_OFF + idx];
        acc += p.x * o.x + p.y * o.y + p.z * o.z + p.w * o.w;
    }
    #pragma unroll
    for (int off = 16; off > 0; off >>= 1)
        acc += __shfl_down(acc, off, 32);
    if (lane == 0) ws[WS_FEAT + i] = fmaxf(acc + b[i], 0.0f);
}

// ---------------- K5: d2[m] = ||som_w[m,:] - feat||^2 via WMMA Gram diagonal ----------------
// Per wave: one 16-row tile, double-buffered in LDS via global_load_async_to_lds_b128
// (ASYNCcnt). A = (w - feat) in WMMA f32 16x16x4 A-layout; passing the same registers
// as A and B gives B = A^T by layout duality, so C = A*A^T and diag(C) = row norms.
__global__ void k_somdist(const float* __restrict__ som, float* __restrict__ ws) {
    __shared__ float sfeat[FEAT];
    __shared__ float stile[2][2][16 * 132];   // [wave][buf][tile], stride 132 = no bank conflicts

    int tid  = threadIdx.x;                 // 64 threads, 2 waves
    int wave = tid >> 5;
    int lane = tid & 31;
    int tile = blockIdx.x * 2 + wave;       // 2048 blocks -> 4096 tiles
    int r0   = tile * 16;

    for (int i = tid; i < FEAT; i += 64) sfeat[i] = ws[WS_FEAT + i];
    __syncthreads();                        // sfeat visible to both waves

    const float* tl0 = &stile[wave][0][0];
    const float* tl1 = &stile[wave][1][0];
    uint32_t lb0 = (uint32_t)(uintptr_t)tl0;   // LDS aperture: low 32 bits = LDS byte offset
    uint32_t lb1 = (uint32_t)(uintptr_t)tl1;

    int hi  = lane >> 4;                    // 0: K,K+1   1: K+2,K+3
    int m   = lane & 15;                    // row within tile (A: M striped over lanes)
    int kb2 = hi * 2;

    // async-issue one 16x128 f32 chunk (16 x b128 per wave, 512B per op)
    auto issue_tile = [&](int k0, uint32_t ldsbase) {
        #pragma unroll
        for (int r = 0; r < 16; ++r) {
            uint64_t ga = (uint64_t)(som + (size_t)(r0 + r) * FEAT + k0 + lane * 4);
            uint32_t la = ldsbase + (uint32_t)((r * 132 + lane * 4) * 4);
            asm volatile("global_load_async_to_lds_b128 %0, %1, off"
                         :: "v"(la), "v"(ga) : "memory");
        }
    };

    v8f c = {};
    auto compute_tile = [&](const float* tlp, int k0) {
        #pragma unroll 8
        for (int kk = 0; kk < 128; kk += 4) {
            v2f w2 = *(const v2f*)&tlp[m * 132 + kk + kb2];
            v2f f2 = *(const v2f*)&sfeat[k0 + kk + kb2];
            v2f a;
            a.x = w2.x - f2.x;
            a.y = w2.y - f2.y;
            c = __builtin_amdgcn_wmma_f32_16x16x4_f32(
                    false, a, false, a, (short)0, c, false, false);
        }
    };

    issue_tile(0,   lb0);
    issue_tile(128, lb1);

    asm volatile("s_wait_asynccnt 0x10" ::: "memory");   // chunk 0 landed
    compute_tile(tl0, 0);
    asm volatile("s_wait_dscnt 0x0" ::: "memory");       // WAR: DS reads of buf0 done
    issue_tile(256, lb0);

    asm volatile("s_wait_asynccnt 0x10" ::: "memory");   // chunk 1 landed
    compute_tile(tl1, 128);
    asm volatile("s_wait_dscnt 0x0" ::: "memory");
    issue_tile(384, lb1);

    asm volatile("s_wait_asynccnt 0x10" ::: "memory");   // chunk 2 landed
    compute_tile(tl0, 256);

    asm volatile("s_wait_asynccnt 0x0" ::: "memory");    // chunk 3 landed
    compute_tile(tl1, 384);

    // diag(C): row r (0..7)  -> VGPR r, lane r     (lanes 0-15 block, M=r,  N=lane)
    //          row r+8       -> VGPR r, lane r+24  (lanes 16-31 block, M=r+8, N=lane-16)
    int q = (lane < 8) ? lane : (lane - 24);
    float d = c[0];
    d = (q == 1) ? c[1] : d;
    d = (q == 2) ? c[2] : d;
    d = (q == 3) ? c[3] : d;
    d = (q == 4) ? c[4] : d;
    d = (q == 5) ? c[5] : d;
    d = (q == 6) ? c[6] : d;
    d = (q == 7) ? c[7] : d;
    int row = (lane < 8) ? lane : (lane - 16);
    if (lane < 8 || lane >= 24) ws[WS_D2 + r0 + row] = d;
}

// ---------------- K6: closed-form SOM loop + final activation map ----------------
// d2_new = (1 - LR*nb)^2 * d2_old per row; 3x (argmin -> rescale), then map = exp(-d2/1024).
__global__ void k_somfinal(float* __restrict__ ws, float* __restrict__ out) {
    __shared__ float sval[1024];
    __shared__ int   sidx[1024];
    int tid = threadIdx.x;
    float* d2 = &ws[WS_D2];

    for (int step = 0; step < 3; ++step) {
        // strided argmin, first-index tie-break (matches jnp.argmax of exp(-d2/...))
        float best = INFINITY;
        int   bi   = 0x7fffffff;
        for (int i = tid; i < MTOT; i += 1024) {
            float v = d2[i];
            if (v < best) { best = v; bi = i; }
        }
        sval[tid] = best; sidx[tid] = bi;
        __syncthreads();
        for (int s = 512; s > 0; s >>= 1) {
            if (tid < s) {
                float v2 = sval[tid + s]; int i2 = sidx[tid + s];
                if (v2 < sval[tid] || (v2 == sval[tid] && i2 < sidx[tid])) {
                    sval[tid] = v2; sidx[tid] = i2;
                }
            }
            __syncthreads();
        }
        int bmu = sidx[0];
        __syncthreads();

        float yb = (float)(bmu >> 8);
        float xb = (float)(bmu & 255);
        for (int i = tid; i < MTOT; i += 1024) {
            float dy = (float)(i >> 8) - yb;
            float dx = (float)(i & 255) - xb;
            float nb = expf(-(dy * dy + dx * dx) * 0.125f);   // /(2*sigma^2), sigma=2
            float s1 = 1.0f - LRATE * nb;
            d2[i] *= s1 * s1;
        }
        __syncthreads();
    }

    const float inv2F = 1.0f / (2.0f * (float)FEAT);
    for (int i = tid; i < MTOT; i += 1024)
        out[i] = expf(-d2[i] * inv2F);
}

extern "C" void kernel_launch(void* const* d_in, const int* in_sizes, int n_in,
                              void* d_out, int out_size, void* d_ws, size_t ws_size,
                              hipStream_t stream) {
    const float* sensory = (const float*)d_in[0];   // [4096,4096]
    const float* W_in    = (const float*)d_in[1];   // [4096,512]
    const float* W_out   = (const float*)d_in[2];   // [512,512]
    const float* proj_W  = (const float*)d_in[3];   // [512,512]
    const float* proj_b  = (const float*)d_in[4];   // [512]
    const float* som_w   = (const float*)d_in[5];   // [65536,512]
    float* ws  = (float*)d_ws;
    float* out = (float*)d_out;

    k_zero<<<(WS_ZERON + 255) / 256, 256, 0, stream>>>(ws, WS_ZERON);
    k_colsum<<<dim3(NN / 256, BATCH / 128), 256, 0, stream>>>(sensory, ws);
    k_gemv_in<<<dim3(FEAT / 256, NN / 256), 256, 0, stream>>>(W_in, ws);
    k_gemv_out<<<dim3(FEAT / 256, FEAT / 128), 256, 0, stream>>>(W_out, ws);
    k_feat<<<FEAT / 8, 256, 0, stream>>>(proj_W, proj_b, ws);
    k_somdist<<<MTOT / 32, 64, 0, stream>>>(som_w, ws);
    k_somfinal<<<1, 1024, 0, stream>>>(ws, out);
}